// MeshPotential_16192026706408
// MI455X (gfx1250) — compile-verified
//
#include <hip/hip_runtime.h>

// ---------------------------------------------------------------------------
// P3M mesh potential for MI455X (gfx1250, wave32).
//   - channels packed pairwise into complex meshes (G(k) real & even)
//   - 256-pt FFT per wave as radix-16^2 using V_WMMA_F32_16X16X4_F32
//   - 6 FFT passes ping-pong bufA<->bufB; each pair (134 MB) is L2-resident
//   - global<->LDS moves use GLOBAL_*_ASYNC_*_LDS_B32 (ASYNCcnt) with
//     8-line z-adjacent tiling so every 8 lanes touch 64 contiguous bytes
// Workspace (d_ws): bufA = 2*NS^3 float2 (268 MB), bufB same => 537 MB.
// ---------------------------------------------------------------------------

#define NSZ   256
#define NS2   (NSZ * NSZ)
#define NS3   ((size_t)NSZ * NSZ * NSZ)
#define WB    272                      // padded per-wave LDS line buffer
#define SMEAR 0.4f
#define TWO_PI  6.28318530717958647692f
#define FOUR_PI 12.5663706143591729539f

typedef __attribute__((ext_vector_type(2))) float v2f;
typedef __attribute__((ext_vector_type(8))) float v8f;

__device__ __forceinline__ v8f wmma4(v2f a, v2f b, v8f c) {
  // D(16x16,f32) = A(16x4,f32) * B(4x16,f32) + C
  return __builtin_amdgcn_wmma_f32_16x16x4_f32(false, a, false, b,
                                               (short)0, c, false, false);
}

// ---- CDNA5 async global<->LDS (ASYNCcnt) ----------------------------------
// Generic->LDS: addrspace(3) offset is the low 32 bits of the generic address.
__device__ __forceinline__ unsigned lds_off(const void* p) {
  return (unsigned)(uintptr_t)p;
}
__device__ __forceinline__ void async_ld_b32(unsigned lds, const float* g) {
  asm volatile("global_load_async_to_lds_b32 %0, %1, off"
               :: "v"(lds), "v"(g) : "memory");
}
__device__ __forceinline__ void async_st_b32(const float* g, unsigned lds) {
  asm volatile("global_store_async_from_lds_b32 %0, %1, off"
               :: "v"(g), "v"(lds) : "memory");
}
__device__ __forceinline__ void wait_async_0() {
  asm volatile("s_wait_asynccnt 0x0" ::: "memory");
}

__device__ __forceinline__ void inv3x3(const float* __restrict__ c,
                                       float inv[9], float* detOut) {
  float a00=c[0],a01=c[1],a02=c[2],a10=c[3],a11=c[4],a12=c[5],a20=c[6],a21=c[7],a22=c[8];
  float det = a00*(a11*a22-a12*a21) - a01*(a10*a22-a12*a20) + a02*(a10*a21-a11*a20);
  float id = 1.0f/det;
  inv[0]=(a11*a22-a12*a21)*id; inv[1]=(a02*a21-a01*a22)*id; inv[2]=(a01*a12-a02*a11)*id;
  inv[3]=(a12*a20-a10*a22)*id; inv[4]=(a00*a22-a02*a20)*id; inv[5]=(a02*a10-a00*a12)*id;
  inv[6]=(a10*a21-a11*a20)*id; inv[7]=(a01*a20-a00*a21)*id; inv[8]=(a00*a11-a01*a10)*id;
  *detOut = det;
}

// Order-4 P3M stencil: 4 wrapped indices + weights per axis (matches reference).
__device__ __forceinline__ void atom_stencil(const float* __restrict__ cell,
                                             const float* __restrict__ pos, int n,
                                             int idx[3][4], float w[3][4]) {
  float inv[9], det;
  inv3x3(cell, inv, &det);
  float p0 = pos[3*n+0], p1 = pos[3*n+1], p2 = pos[3*n+2];
#pragma unroll
  for (int d = 0; d < 3; ++d) {
    float pr = (float)NSZ * (p0*inv[0+d] + p1*inv[3+d] + p2*inv[6+d]);
    float f  = floorf(pr);
    int   i0 = (int)f;
    float x  = pr - (f + 0.5f);
    float x2 = x*x, x3 = x2*x;
    w[d][0] = (1.f  - 6.f*x  + 12.f*x2 - 8.f*x3)  * (1.f/48.f);
    w[d][1] = (23.f - 30.f*x - 12.f*x2 + 24.f*x3) * (1.f/48.f);
    w[d][2] = (23.f + 30.f*x - 12.f*x2 - 24.f*x3) * (1.f/48.f);
    w[d][3] = (1.f  + 6.f*x  + 12.f*x2 + 8.f*x3)  * (1.f/48.f);
#pragma unroll
    for (int j = 0; j < 4; ++j) idx[d][j] = (i0 + j - 1 + NSZ) & (NSZ - 1);
  }
}

// ---------------------------------------------------------------------------
// Scatter: mesh pair 0 = (ch0 -> re, ch1 -> im), pair 1 = (ch2 -> re, ch3 -> im)
// ---------------------------------------------------------------------------
__global__ void scatter_kernel(const float* __restrict__ cell,
                               const float* __restrict__ pos,
                               const float* __restrict__ q,
                               float2* __restrict__ mesh, int natoms) {
  int n = blockIdx.x * blockDim.x + threadIdx.x;
  if (n >= natoms) return;
  int idx[3][4]; float w[3][4];
  atom_stencil(cell, pos, n, idx, w);
  float q0 = q[4*n+0], q1 = q[4*n+1], q2 = q[4*n+2], q3 = q[4*n+3];
#pragma unroll
  for (int a = 0; a < 4; ++a) {
    float wa = w[0][a]; size_t ia = (size_t)idx[0][a] * NS2;
#pragma unroll
    for (int b = 0; b < 4; ++b) {
      float wab = wa * w[1][b]; size_t iab = ia + (size_t)idx[1][b] * NSZ;
#pragma unroll
      for (int c3 = 0; c3 < 4; ++c3) {
        float w3 = wab * w[2][c3];
        size_t f = iab + (size_t)idx[2][c3];
        if (q0 != 0.f) atomicAdd(&mesh[f].x,       w3 * q0);
        if (q1 != 0.f) atomicAdd(&mesh[f].y,       w3 * q1);
        if (q2 != 0.f) atomicAdd(&mesh[NS3 + f].x, w3 * q2);
        if (q3 != 0.f) atomicAdd(&mesh[NS3 + f].y, w3 * q3);
      }
    }
  }
}

// ---------------------------------------------------------------------------
// 256-pt complex FFT per wave, radix-16 x radix-16, WMMA f32 16x16x4.
//   n = 16*n1 + n2 ; k = k1 + 16*k2
//   stage1: Y[k1][n2]  = sum_n1 F16[k1][n1] * X[n1][n2]      (F * X)
//   twiddle: Y'[k1][n2] = Y * exp(sgn*2pi*i*k1*n2/256)
//   stage2: Z[k1][k2]  = sum_n2 Y'[k1][n2] * F16[n2][k2]     (Y' * F)
// Layouts: A(16x4): lane l -> M=l&15; VGPR0/1 hold K = 2*(l>>4) +0/1 of chunk.
//          B(4x16): mirrored (lane -> N=l&15, VGPRs hold chunk K rows).
//          C/D: VGPR r, lane l -> (M = r + 8*(l>>4), N = l&15).
// Complex cross-terms via negated table values (f32 WMMA has no A/B neg).
// ---------------------------------------------------------------------------
__device__ __forceinline__ void wave_fft256(float* sre, float* sim,
                                            const float* tcos, const float* tsin) {
  const int lane = threadIdx.x & 31;
  const int ml   = lane & 15;
  const int hi   = lane >> 4;

  // ---- stage 1: Y = F16 * X (X linear at sre[16*n1 + n2]) ----
  v8f yr = {}; v8f yi = {};
#pragma unroll
  for (int kk = 0; kk < 4; ++kk) {
    const int k0 = 4*kk + 2*hi;
    const int k1 = k0 + 1;
    v2f ar, ai, br, bi;
    ar.x = tcos[(ml*k0*16) & 255]; ar.y = tcos[(ml*k1*16) & 255];
    ai.x = tsin[(ml*k0*16) & 255]; ai.y = tsin[(ml*k1*16) & 255];
    br.x = sre[16*k0 + ml];        br.y = sre[16*k1 + ml];
    bi.x = sim[16*k0 + ml];        bi.y = sim[16*k1 + ml];
    v2f ain = -ai;
    yr = wmma4(ar,  br, yr);   // Re += Fr*Xr
    yr = wmma4(ain, bi, yr);   // Re -= Fi*Xi
    yi = wmma4(ar,  bi, yi);   // Im += Fr*Xi
    yi = wmma4(ai,  br, yi);   // Im += Fi*Xr
  }
  __syncthreads();

  // ---- twiddle + transpose-store Y' at padded stride 17 ----
#pragma unroll
  for (int r = 0; r < 8; ++r) {
    const int m = r + 8*hi;                 // k1
    const float c = tcos[m*ml];             // exp(sgn*2pi*i*k1*n2/256)
    const float s = tsin[m*ml];
    const float vr = yr[r], vi = yi[r];
    sre[17*m + ml] = vr*c - vi*s;
    sim[17*m + ml] = vr*s + vi*c;
  }
  __syncthreads();

  // ---- stage 2: Z = Y' * F16 ----
  v8f zr = {}; v8f zi = {};
#pragma unroll
  for (int kk = 0; kk < 4; ++kk) {
    const int k0 = 4*kk + 2*hi;
    const int k1 = k0 + 1;
    v2f ar, ai, br, bi;
    ar.x = sre[17*ml + k0];        ar.y = sre[17*ml + k1];
    ai.x = sim[17*ml + k0];        ai.y = sim[17*ml + k1];
    br.x = tcos[(k0*ml*16) & 255]; br.y = tcos[(k1*ml*16) & 255];
    bi.x = tsin[(k0*ml*16) & 255]; bi.y = tsin[(k1*ml*16) & 255];
    v2f bin = -bi;
    zr = wmma4(ar, br,  zr);
    zr = wmma4(ai, bin, zr);
    zi = wmma4(ar, bi,  zi);
    zi = wmma4(ai, br,  zi);
  }
  __syncthreads();

  // ---- store Z at bit-natural output index k = k1 + 16*k2 ----
#pragma unroll
  for (int r = 0; r < 8; ++r) {
    const int m = r + 8*hi;       // k1
    const int k = m + 16*ml;      // k2 = ml
    sre[k] = zr[r];
    sim[k] = zi[r];
  }
  __syncthreads();
}

// ---------------------------------------------------------------------------
// One FFT pass along `axis`. Each block owns 8 lines chosen adjacent in z
// (the memory-contiguous axis) so the async tile moves are cacheline-dense:
//   axis==2 : wave wid owns line (blockIdx.x*8+wid); its elements are
//             contiguous, lanes stream 64B-dense segments.
//   axis==1 : lines (x=maj, z=zb..zb+7); elem t of line j at
//             maj*NS2 + t*NS + zb + j  -> 8 lanes (j=0..7) cover 64 B.
//   axis==0 : lines (y=maj, z=zb..zb+7); elem t at t*NS2 + maj*NS + zb + j.
// Loads/stores go straight global<->LDS via ASYNCcnt ops (no VGPR staging).
// ---------------------------------------------------------------------------
__global__ __launch_bounds__(256)
void fft_pass_kernel(const float2* __restrict__ src, float2* __restrict__ dst,
                     int axis, float sgn) {
  __shared__ float lre[8 * WB];
  __shared__ float lim[8 * WB];
  __shared__ float tcos[256];
  __shared__ float tsin[256];

  const int tid = threadIdx.x;
  {
    const float ang = sgn * (TWO_PI / 256.0f) * (float)tid;
    tcos[tid] = cosf(ang);
    tsin[tid] = sinf(ang);
  }

  const int wid  = tid >> 5;
  const int lane = tid & 31;
  const float2* s = src + (size_t)blockIdx.y * NS3;   // channel-pair select
  float2*       d = dst + (size_t)blockIdx.y * NS3;

  size_t base, stride;
  int j, tq;                     // j = LDS line this thread moves, tq = row phase
  if (axis == 2) {
    base = ((size_t)blockIdx.x * 8 + wid) * NSZ;
    stride = 1;
    j = wid; tq = lane;
  } else {
    const int maj = blockIdx.x >> 5;
    const int zb  = (blockIdx.x & 31) * 8;
    if (axis == 1) { base = (size_t)maj * NS2 + (size_t)zb; stride = NSZ; }
    else           { base = (size_t)maj * NSZ + (size_t)zb; stride = NS2; }
    j = tid & 7; tq = tid >> 3;
    base += (size_t)j;           // contiguous z offset within the 8-line group
  }

  // ---- async global -> LDS (deinterleave re/im into separate arrays) ----
#pragma unroll
  for (int r = 0; r < 8; ++r) {
    const int t = r * 32 + tq;
    const float* gp = (const float*)(s + base + (size_t)t * stride);
    async_ld_b32(lds_off(&lre[j * WB + t]), gp);
    async_ld_b32(lds_off(&lim[j * WB + t]), gp + 1);
  }
  wait_async_0();
  __syncthreads();               // tile writes cross waves; also table visibility

  wave_fft256(lre + wid * WB, lim + wid * WB, tcos, tsin);

  // ---- async LDS -> global (mirror mapping) ----
#pragma unroll
  for (int r = 0; r < 8; ++r) {
    const int t = r * 32 + tq;
    const float* gp = (const float*)(d + base + (size_t)t * stride);
    async_st_b32(gp,     lds_off(&lre[j * WB + t]));
    async_st_b32(gp + 1, lds_off(&lim[j * WB + t]));
  }
  // ASYNCcnt drains via the implicit wait-idle at S_ENDPGM.
}

// ---------------------------------------------------------------------------
// Pointwise Coulomb kernel: G(k) = 4*pi*exp(-0.5*sigma^2*k^2)/k^2, G(0)=0,
// folded with the 1/det(cell) of the final normalization. Applied to both pairs.
// ---------------------------------------------------------------------------
__global__ __launch_bounds__(256)
void greens_kernel(const float* __restrict__ cell, float2* __restrict__ spec) {
  size_t i = (size_t)blockIdx.x * 256 + threadIdx.x;
  int iz = (int)(i & (NSZ - 1));
  int iy = (int)((i >> 8) & (NSZ - 1));
  int ix = (int)(i >> 16);

  float inv[9], det;
  inv3x3(cell, inv, &det);

  float fx = (ix < NSZ/2) ? (float)ix : (float)(ix - NSZ);
  float fy = (iy < NSZ/2) ? (float)iy : (float)(iy - NSZ);
  float fz = (iz < NSZ/2) ? (float)iz : (float)(iz - NSZ);

  // k = 2*pi * (fx,fy,fz) @ inv_cell^T  -> component c uses row c of inv_cell
  float kx = TWO_PI * (fx*inv[0] + fy*inv[1] + fz*inv[2]);
  float ky = TWO_PI * (fx*inv[3] + fy*inv[4] + fz*inv[5]);
  float kz = TWO_PI * (fx*inv[6] + fy*inv[7] + fz*inv[8]);
  float ksq = kx*kx + ky*ky + kz*kz;

  float g = 0.0f;
  if (i != 0) g = FOUR_PI * expf(-0.5f * SMEAR * SMEAR * ksq) / ksq;
  g /= det;

  float2 v0 = spec[i];       v0.x *= g; v0.y *= g; spec[i]       = v0;
  float2 v1 = spec[NS3 + i]; v1.x *= g; v1.y *= g; spec[NS3 + i] = v1;
}

// ---------------------------------------------------------------------------
// Gather: pot channels = (pair0.re, pair0.im, pair1.re, pair1.im)
// ---------------------------------------------------------------------------
__global__ void gather_kernel(const float* __restrict__ cell,
                              const float* __restrict__ pos,
                              const float2* __restrict__ mesh,
                              float* __restrict__ out, int natoms) {
  int n = blockIdx.x * blockDim.x + threadIdx.x;
  if (n >= natoms) return;
  int idx[3][4]; float w[3][4];
  atom_stencil(cell, pos, n, idx, w);
  float a0 = 0.f, a1 = 0.f, a2 = 0.f, a3 = 0.f;
#pragma unroll
  for (int a = 0; a < 4; ++a) {
    float wa = w[0][a]; size_t ia = (size_t)idx[0][a] * NS2;
#pragma unroll
    for (int b = 0; b < 4; ++b) {
      float wab = wa * w[1][b]; size_t iab = ia + (size_t)idx[1][b] * NSZ;
#pragma unroll
      for (int c3 = 0; c3 < 4; ++c3) {
        float w3 = wab * w[2][c3];
        size_t f = iab + (size_t)idx[2][c3];
        float2 v0 = mesh[f];
        float2 v1 = mesh[NS3 + f];
        a0 += w3 * v0.x; a1 += w3 * v0.y;
        a2 += w3 * v1.x; a3 += w3 * v1.y;
      }
    }
  }
  out[4*n+0] = a0; out[4*n+1] = a1; out[4*n+2] = a2; out[4*n+3] = a3;
}

// ---------------------------------------------------------------------------
extern "C" void kernel_launch(void* const* d_in, const int* in_sizes, int n_in,
                              void* d_out, int out_size, void* d_ws, size_t ws_size,
                              hipStream_t stream) {
  (void)n_in; (void)out_size; (void)ws_size;
  const float* cell      = (const float*)d_in[0];   // (3,3) f32
  const float* positions = (const float*)d_in[1];   // (N,3) f32
  const float* charges   = (const float*)d_in[2];   // (N,4) f32 one-hot
  const int natoms = in_sizes[1] / 3;

  float2* bufA = reinterpret_cast<float2*>(d_ws);   // 2 pairs * NS^3 complex
  float2* bufB = bufA + (size_t)2 * NS3;

  // zero rho meshes (both pairs), then scatter atoms
  hipMemsetAsync(bufA, 0, (size_t)2 * NS3 * sizeof(float2), stream);
  scatter_kernel<<<(natoms + 127) / 128, 128, 0, stream>>>(cell, positions, charges,
                                                           bufA, natoms);

  const dim3 fgrid(NS2 / 8, 2);                     // 8 lines/block, 2 channel-pairs
  // forward FFT (raw, e^{-i}) : A -> B -> A -> B
  fft_pass_kernel<<<fgrid, 256, 0, stream>>>(bufA, bufB, 2, -1.0f);
  fft_pass_kernel<<<fgrid, 256, 0, stream>>>(bufB, bufA, 1, -1.0f);
  fft_pass_kernel<<<fgrid, 256, 0, stream>>>(bufA, bufB, 0, -1.0f);
  // multiply by G(k)/det in-place on spectrum (both pairs)
  greens_kernel<<<(unsigned)(NS3 / 256), 256, 0, stream>>>(cell, bufB);
  // inverse FFT (raw, e^{+i}, norm='forward' => no 1/N^3) : B -> A -> B -> A
  fft_pass_kernel<<<fgrid, 256, 0, stream>>>(bufB, bufA, 2, +1.0f);
  fft_pass_kernel<<<fgrid, 256, 0, stream>>>(bufA, bufB, 1, +1.0f);
  fft_pass_kernel<<<fgrid, 256, 0, stream>>>(bufB, bufA, 0, +1.0f);

  gather_kernel<<<(natoms + 127) / 128, 128, 0, stream>>>(cell, positions, bufA,
                                                          (float*)d_out, natoms);
}